// GaussianBridge_1236950581708
// MI455X (gfx1250) — compile-verified
//
#include <hip/hip_runtime.h>

typedef __attribute__((ext_vector_type(2))) float v2f;
typedef __attribute__((ext_vector_type(4))) float v4f;
typedef __attribute__((ext_vector_type(8))) float v8f;

#define LDS_STRIDE 520   // 512 + 8 pad: rows k and k+2 hit disjoint bank sets for the two lane halves
#define MAXK 16

__device__ __forceinline__ float softplus_f(float x) {
    // jax.nn.softplus = max(x,0) + log1p(exp(-|x|))
    return fmaxf(x, 0.0f) + log1pf(expf(-fabsf(x)));
}

// ---------------------------------------------------------------------------
// diffusion = 1.0 everywhere: pure streaming b128 nontemporal fill
// ---------------------------------------------------------------------------
__global__ __launch_bounds__(256) void fill_ones_kernel(float* __restrict__ p, size_t n4) {
    size_t i = (size_t)blockIdx.x * blockDim.x + threadIdx.x;
    if (i < n4) {
        v4f one = {1.0f, 1.0f, 1.0f, 1.0f};
        __builtin_nontemporal_store(one, (v4f*)(p + 4 * i));
    }
}

// ---------------------------------------------------------------------------
// drift kernel: spline-as-GEMM on f32 WMMA (16x16x4), K = 15 knots padded to 16
// one block = 16 rows x 512 cols; 8 waves x 4 col-tiles each
// ---------------------------------------------------------------------------
__global__ __launch_bounds__(256) void bridge_kernel(
    const float* __restrict__ z, const float* __restrict__ t,
    const float* __restrict__ fixed_mu, const float* __restrict__ fixed_times,
    const float* __restrict__ ctrl_times, const float* __restrict__ mu_ctrl,
    const float* __restrict__ gamma_fixed, const float* __restrict__ gamma_raw_ctrl,
    float* __restrict__ drift_out,
    int Bn, int D, int NF, int NC)
{
    __shared__ float P_lds[16 * LDS_STRIDE];
    __shared__ float times_s[MAXK];
    __shared__ float g_s[MAXK];
    __shared__ int   src_s[MAXK];
    __shared__ float c1_s[16];

    const int K   = NF + NC;          // 15 for the reference shapes
    const int tid = threadIdx.x;

    // 1) zero knot tile (pad rows must be exact 0 so 0*P contributions are 0, never NaN)
    for (int e = tid; e < 16 * LDS_STRIDE; e += (int)blockDim.x) P_lds[e] = 0.0f;

    // 2) thread 0: selection-sort the <=15 knot times, apply softplus to gamma controls
    if (tid == 0) {
        float tt[MAXK]; int si[MAXK];
        for (int i = 0; i < NF; ++i) { tt[i] = fixed_times[i]; si[i] = i; }
        for (int i = 0; i < NC; ++i) { tt[NF + i] = ctrl_times[i]; si[NF + i] = NF + i; }
        for (int i = 0; i < K - 1; ++i) {                 // stable selection sort
            int mn = i;
            for (int j = i + 1; j < K; ++j) if (tt[j] < tt[mn]) mn = j;
            float tv = tt[i]; tt[i] = tt[mn]; tt[mn] = tv;
            int   iv = si[i]; si[i] = si[mn]; si[mn] = iv;
        }
        for (int i = 0; i < K; ++i) {
            times_s[i] = tt[i];
            src_s[i]   = si[i];
            g_s[i]     = (si[i] < NF) ? gamma_fixed[si[i]]
                                      : softplus_f(gamma_raw_ctrl[si[i] - NF]);
        }
        for (int i = K; i < MAXK; ++i) { times_s[i] = 3.0e38f; src_s[i] = -1; g_s[i] = 0.0f; }
    }
    __syncthreads();

    // 3) cooperatively stage sorted knot matrix P[K x D] into LDS (L2-hot reads)
    for (int e = tid; e < K * D; e += (int)blockDim.x) {
        int k = e / D, col = e - k * D;
        int s = src_s[k];
        float v = (s < NF) ? fixed_mu[s * D + col] : mu_ctrl[(s - NF) * D + col];
        P_lds[k * LDS_STRIDE + col] = v;
    }

    // 4) per-row spline interval params; lane L handles row (L & 15) of this 16-row tile
    const int lane = tid & 31;
    const int hl   = lane & 15;     // 0..15
    const int hi16 = lane >> 4;     // 0 (lanes 0-15) or 1 (lanes 16-31)
    const int row_base = blockIdx.x * 16;

    float t_r = t[row_base + hl];
    int cnt = 0;                                    // searchsorted(times, t, 'right')
    for (int k = 0; k < K; ++k) cnt += (times_s[k] <= t_r) ? 1 : 0;
    int right = cnt < 1 ? 1 : (cnt > K - 1 ? K - 1 : cnt);
    int left  = right - 1;
    bool lo = (t_r <= times_s[0]);
    bool hi = (t_r >= times_s[K - 1]);
    float alpha, dscale;
    if (lo)      { left = 0;     right = 0;     alpha = 0.0f; dscale = 0.0f; }
    else if (hi) { left = K - 1; right = K - 1; alpha = 0.0f; dscale = 0.0f; }
    else {
        float tl = times_s[left], tr = times_s[right];
        dscale = 1.0f / (tr - tl);
        alpha  = (t_r - tl) * dscale;
    }
    // gamma spline (scalar per row) and combined coefficient c1 = dg/g + 0.5/g^2
    float gl = g_s[left], gr = g_s[right];
    float gamma  = (1.0f - alpha) * gl + alpha * gr;
    float dgamma = (lo || hi) ? 0.0f : (gr - gl) * dscale;
    gamma = fmaxf(gamma, 1e-6f);
    float c1 = dgamma / gamma + 0.5f / (gamma * gamma);
    if (tid < 16) c1_s[tid] = c1;    // wave 0 lanes 0-15 cover rows 0-15

    // 5) A fragments (16x4 f32 layout: lanes 0-15 carry K = 4j+{0,1}, lanes 16-31 K = 4j+{2,3})
    const int koff = hi16 * 2;
    v2f Amu[4], Ad[4];
    #pragma unroll
    for (int j = 0; j < 4; ++j) {
        int k0 = 4 * j + koff;
        int k1 = k0 + 1;
        float w0 = (k0 == left) ? (1.0f - alpha) : ((k0 == right) ? alpha  : 0.0f);
        float w1 = (k1 == left) ? (1.0f - alpha) : ((k1 == right) ? alpha  : 0.0f);
        float d0 = (k0 == left) ? (-dscale)      : ((k0 == right) ? dscale : 0.0f);
        float d1 = (k1 == left) ? (-dscale)      : ((k1 == right) ? dscale : 0.0f);
        Amu[j].x = w0; Amu[j].y = w1;
        Ad[j].x  = d0; Ad[j].y  = d1;
    }
    __syncthreads();   // P_lds staged + c1_s visible

    // hoist c1 broadcasts out of the column-tile loop (same rows for all tiles)
    float c1r[8];
    #pragma unroll
    for (int r = 0; r < 8; ++r) c1r[r] = c1_s[r + (hi16 << 3)];

    // 6) per-wave loop over 16-col tiles: mu/dmu via chained V_WMMA_F32_16X16X4_F32
    const int wave  = tid >> 5;
    const int ntile = D >> 4;
    const unsigned uD = (unsigned)D;
    for (int ct = wave; ct < ntile; ct += 8) {
        int col0 = ct << 4;
        v8f accm = {0.f,0.f,0.f,0.f,0.f,0.f,0.f,0.f};
        v8f accd = {0.f,0.f,0.f,0.f,0.f,0.f,0.f,0.f};
        #pragma unroll
        for (int j = 0; j < 4; ++j) {
            int k0 = 4 * j + koff;
            v2f Bf;                                  // B 4x16: rows striped across lane halves
            Bf.x = P_lds[k0       * LDS_STRIDE + col0 + hl];
            Bf.y = P_lds[(k0 + 1) * LDS_STRIDE + col0 + hl];
            accm = __builtin_amdgcn_wmma_f32_16x16x4_f32(
                false, Amu[j], false, Bf, (short)0, accm, false, false);
            accd = __builtin_amdgcn_wmma_f32_16x16x4_f32(
                false, Ad[j],  false, Bf, (short)0, accd, false, false);
        }
        // epilogue: C/D layout -> lane owns (row r + 8*hi16, col hl) for acc index r.
        // 32-bit base offset + constant r*D deltas => saddr + voffset + imm addressing
        unsigned base = (unsigned)(row_base + (hi16 << 3)) * uD + (unsigned)(col0 + hl);
        #pragma unroll
        for (int r = 0; r < 8; ++r) {
            unsigned e = base + (unsigned)r * uD;
            float zv = __builtin_nontemporal_load(z + e);
            float drift = fmaf(c1r[r], zv - accm[r], accd[r]);
            __builtin_nontemporal_store(drift, drift_out + e);
        }
    }
}

extern "C" void kernel_launch(void* const* d_in, const int* in_sizes, int n_in,
                              void* d_out, int out_size, void* d_ws, size_t ws_size,
                              hipStream_t stream) {
    const float* z   = (const float*)d_in[0];
    const float* t   = (const float*)d_in[1];
    const float* fpm = (const float*)d_in[2];
    const float* ftm = (const float*)d_in[3];
    const float* ctm = (const float*)d_in[4];
    const float* mc  = (const float*)d_in[5];
    const float* gf  = (const float*)d_in[6];
    const float* grc = (const float*)d_in[7];

    const int BD = in_sizes[0];        // B*D = 33554432
    const int Bn = in_sizes[1];        // 65536
    const int D  = BD / Bn;            // 512
    const int NF = in_sizes[3];        // 5
    const int NC = in_sizes[4];        // 10

    float* drift = (float*)d_out;
    float* diff  = drift + (size_t)BD;

    // diffusion half: b128 nontemporal fill of 1.0f
    size_t n4 = (size_t)BD / 4;
    fill_ones_kernel<<<(unsigned)((n4 + 255) / 256), 256, 0, stream>>>(diff, n4);

    // drift half: one block per 16 rows
    bridge_kernel<<<(unsigned)(Bn / 16), 256, 0, stream>>>(
        z, t, fpm, ftm, ctm, mc, gf, grc, drift, Bn, D, NF, NC);
}